// PredictYolov1_33208687132987
// MI455X (gfx1250) — compile-verified
//
#include <hip/hip_runtime.h>
#include <stdint.h>

#define GG 7
#define NBB 2
#define NCLS 20
#define NCELL (GG * GG)          // 49
#define NBOX (NCELL * NBB)       // 98
#define CH (NBB * 5 + NCLS)      // 30
#define TILEF (NCELL * CH)       // 1470 floats = 5880 bytes per image
#define BLOCK 128
#define THR_CONF 0.5f
#define THR_NMS 0.3f

typedef __attribute__((ext_vector_type(4))) unsigned int v4u;
typedef __attribute__((ext_vector_type(4))) int v4i;
typedef __attribute__((ext_vector_type(8))) int v8i;

__device__ __forceinline__ float sigmoidf_(float x) {
  return 1.0f / (1.0f + __expf(-x));
}

__global__ __launch_bounds__(BLOCK)
void yolo_decode_nms_kernel(const float* __restrict__ p,
                            float* __restrict__ out_boxes,
                            float* __restrict__ out_scores,
                            float* __restrict__ out_labels,
                            float* __restrict__ out_keep) {
  __shared__ float tile[TILEF];         // staged input image tile
  __shared__ float s_score[NBOX];
  __shared__ int   s_label[NBOX];
  __shared__ float s_box[NBOX][4];
  __shared__ float o_box[NBOX][4];      // sorted (descending score, stable)
  __shared__ float o_score[NBOX];
  __shared__ int   o_label[NBOX];
  __shared__ int   o_valid[NBOX];
  __shared__ int   o_orig[NBOX];
  __shared__ int   o_keep[NBOX];

  const int img = blockIdx.x;
  const int t = threadIdx.x;
  const float* src = p + (size_t)img * TILEF;

  // ---- Stage the per-image tile into LDS with the Tensor Data Mover ----
#if __has_builtin(__builtin_amdgcn_tensor_load_to_lds)
  if (t < 32) {  // uniform over wave 0 (wave32); TDM ignores EXEC, issues once per wave
    unsigned long long ga = (unsigned long long)(uintptr_t)src;
    v4u g0;
    g0.x = 1u;                                                 // count=1, user descriptor
    g0.y = (unsigned)(uintptr_t)(&tile[0]);                    // lds_addr (bytes)
    g0.z = (unsigned)ga;                                       // global_addr[31:0]
    g0.w = (unsigned)((ga >> 32) & 0x01FFFFFFu) | (2u << 30);  // addr[56:32] | type=2
    v8i g1;
    g1[0] = (int)(2u << 16);                                   // data_size = 4 bytes
    g1[1] = (int)(((unsigned)TILEF & 0xFFFFu) << 16);          // tensor_dim0[15:0]
    g1[2] = (int)(((unsigned)TILEF >> 16) | (1u << 16));       // dim0[31:16] | tensor_dim1=1
    g1[3] = (int)(((unsigned)TILEF & 0xFFFFu) << 16);          // tile_dim0 = 1470 (1-D tile)
    g1[4] = 0;                                                 // tile_dim1=0, tile_dim2=0
    g1[5] = (int)TILEF;                                        // tensor_dim0_stride
    g1[6] = 0;
    g1[7] = 0;
    v4i gz = {0, 0, 0, 0};
#if defined(__clang_major__) && (__clang_major__ >= 23)
    v8i gz8 = {0, 0, 0, 0, 0, 0, 0, 0};
    __builtin_amdgcn_tensor_load_to_lds(g0, g1, gz, gz, gz8, 0);
#else
    __builtin_amdgcn_tensor_load_to_lds(g0, g1, gz, gz, 0);
#endif
  }
  __builtin_amdgcn_s_wait_tensorcnt(0);  // no-op for waves that didn't issue
#else
  for (int i = t; i < TILEF; i += BLOCK) tile[i] = src[i];
#endif
  __syncthreads();

  // ---- Decode: thread t owns box t (cell = t/2, bbox = t%2) ----
  float bx1 = 0.f, by1 = 0.f, bx2 = 0.f, by2 = 0.f, areaT = 0.f;
  int lblT = 0;
  if (t < NBOX) {
    int cell = t >> 1, b = t & 1;
    int r = cell / GG, c = cell - r * GG;
    const float* cb = &tile[cell * CH];
    float tx = cb[4 * b + 0], ty = cb[4 * b + 1];
    float tw = cb[4 * b + 2], th = cb[4 * b + 3];
    float gx = (sigmoidf_(tx) + (float)c) / (float)GG;
    float gy = (sigmoidf_(ty) + (float)r) / (float)GG;
    float x1 = gx - tw * 0.5f, y1 = gy - th * 0.5f;
    float x2 = gx + tw * 0.5f, y2 = gy + th * 0.5f;
    float conf = sigmoidf_(cb[8 + b]);
    float best = cb[10];
    int bi = 0;
#pragma unroll
    for (int k = 1; k < NCLS; ++k) {   // first-max, matches jnp.argmax
      float v = cb[10 + k];
      if (v > best) { best = v; bi = k; }
    }
    int label = bi + 1;
    s_score[t] = conf;
    s_label[t] = label;
    s_box[t][0] = x1; s_box[t][1] = y1; s_box[t][2] = x2; s_box[t][3] = y2;
    size_t ob = (size_t)img * NBOX + t;
    out_boxes[ob * 4 + 0] = x1; out_boxes[ob * 4 + 1] = y1;
    out_boxes[ob * 4 + 2] = x2; out_boxes[ob * 4 + 3] = y2;
    out_scores[ob] = conf;
    out_labels[ob] = (float)label;
  }
  __syncthreads();

  // ---- Stable descending rank (== jnp.argsort(-scores)) + scatter ----
  if (t < NBOX) {
    float st = s_score[t];
    int rank = 0;
    for (int j = 0; j < NBOX; ++j) {
      float sj = s_score[j];
      rank += (int)(sj > st) + (int)((sj == st) && (j < t));
    }
    o_score[rank] = st;
    o_label[rank] = s_label[t];
    o_valid[rank] = (st > THR_CONF) ? 1 : 0;
    o_orig[rank] = t;
    o_box[rank][0] = s_box[t][0]; o_box[rank][1] = s_box[t][1];
    o_box[rank][2] = s_box[t][2]; o_box[rank][3] = s_box[t][3];
  }
  __syncthreads();

  if (t < NBOX) {
    bx1 = o_box[t][0]; by1 = o_box[t][1];
    bx2 = o_box[t][2]; by2 = o_box[t][3];
    areaT = fmaxf(bx2 - bx1, 0.f) * fmaxf(by2 - by1, 0.f);
    lblT = o_label[t];
  }

  // ---- Greedy NMS over sorted order (98-step serial chain) ----
  bool sup = false;
  for (int pos = 0; pos < NBOX; ++pos) {
    if (t == pos) o_keep[pos] = (o_valid[pos] && !sup) ? 1 : 0;
    __syncthreads();
    if (t < NBOX && t > pos && o_keep[pos] && (o_label[pos] == lblT)) {
      float px1 = o_box[pos][0], py1 = o_box[pos][1];
      float px2 = o_box[pos][2], py2 = o_box[pos][3];
      float ix1 = fmaxf(px1, bx1), iy1 = fmaxf(py1, by1);
      float ix2 = fminf(px2, bx2), iy2 = fminf(py2, by2);
      float iw = fmaxf(ix2 - ix1, 0.f), ih = fmaxf(iy2 - iy1, 0.f);
      float inter = iw * ih;
      float pa = fmaxf(px2 - px1, 0.f) * fmaxf(py2 - py1, 0.f);
      float iou = inter / fmaxf(pa + areaT - inter, 1e-9f);
      if (iou > THR_NMS) sup = true;
    }
    __syncthreads();
  }

  // ---- Scatter keep back to original order ----
  if (t < NBOX) {
    out_keep[(size_t)img * NBOX + o_orig[t]] = (float)o_keep[t];
  }
}

extern "C" void kernel_launch(void* const* d_in, const int* in_sizes, int n_in,
                              void* d_out, int out_size, void* d_ws, size_t ws_size,
                              hipStream_t stream) {
  (void)n_in; (void)out_size; (void)d_ws; (void)ws_size;
  const float* p = (const float*)d_in[0];
  int B = in_sizes[0] / TILEF;               // 8192
  float* out = (float*)d_out;
  size_t nb = (size_t)B * NBOX;
  float* ob = out;                           // boxes: B*98*4
  float* os = out + nb * 4;                  // scores: B*98
  float* ol = os + nb;                       // labels: B*98 (as float)
  float* ok = ol + nb;                       // keep:   B*98 (as float 0/1)
  yolo_decode_nms_kernel<<<B, BLOCK, 0, stream>>>(p, ob, os, ol, ok);
}